// WITLayer_36739150250539
// MI455X (gfx1250) — compile-verified
//
#include <hip/hip_runtime.h>
#include <stdint.h>

// Inverse 2x2 Haar synthesis, bandwidth-bound (~403 MB traffic, floor ~17us @ 23.3 TB/s).
// CDNA5 path: TDM (tensor_load_to_lds) DMAs the 4 subband rows into LDS per block,
// butterfly in VALU, interleaved 128-bit non-temporal stores to the 2x-upsampled output.

typedef float        f32x4 __attribute__((ext_vector_type(4)));
typedef unsigned int u32x4 __attribute__((ext_vector_type(4)));
typedef int          i32x4 __attribute__((ext_vector_type(4)));
typedef int          i32x8 __attribute__((ext_vector_type(8)));

#define B_  16
#define HH  512
#define WW  512

__global__ __launch_bounds__(256) void iwt_haar_tdm(const float* __restrict__ in,
                                                    float* __restrict__ out) {
    __shared__ float tile[4 * WW];      // 8 KB: cA row | cH row | cV row | cD row

    const int blk = blockIdx.x;                 // 0 .. 16*3*512-1
    const int h   = blk & (HH - 1);
    const int c   = (blk >> 9) % 3;
    const int b   = blk / (3 * HH);

    // Flat LDS address = {SHARED_BASE, offset}; addr[31:0] is the LDS byte offset
    // (aperture mapping, ISA 10.2). The ptrtoint also makes `tile` escape so the
    // compiler cannot fold the LDS reads below to undef (the TDM write is opaque).
    const uint32_t lds_base = (uint32_t)(uintptr_t)(void*)tile;

    // ---- wave 0 issues ONE TDM 2D load: 4 planes x 512 floats -> LDS ----
    // Tile addressing (ISA 8.2): Maddr = global_addr + 4*(X + Y*tensor_dim0_stride)
    // with Y = plane index (stride = H*W elements) and X = column. LDS fills
    // contiguously: plane s at tile[s*512 .. s*512+511].
    if (threadIdx.x < 32) {
        const uint64_t ga = (uint64_t)(uintptr_t)(
            in + ((((uint64_t)b * 12 + (uint64_t)c * 4) * HH + h) * WW));

        u32x4 g0;
        g0[0] = 1u;                                        // [1:0] count=1 (valid user D#)
        g0[1] = lds_base;                                  // [63:32] lds_addr (bytes)
        g0[2] = (uint32_t)(ga & 0xFFFFFFFFu);              // [95:64] global_addr lo
        g0[3] = (uint32_t)((ga >> 32) & 0x01FFFFFFu)       // [120:96] global_addr hi
              | (2u << 30);                                // [127:126] type=2 ("image")

        i32x8 g1;
        g1[0] = (int)(2u << 16);        // [15:0] wg_mask=0 (not in cluster), [17:16] data_size=2 (4B)
        g1[1] = (int)((unsigned)WW << 16);   // [47:32] atomic_barrier_addr=0, [63:48] tensor_dim0.lo = 512
        g1[2] = (int)(4u << 16);        // [79:64] tensor_dim0.hi=0, [95:80] tensor_dim1.lo = 4 planes
        g1[3] = (int)((unsigned)WW << 16);   // [111:96] tensor_dim1.hi=0, [127:112] tile_dim0 = 512
        g1[4] = 4;                      // [143:128] tile_dim1 = 4, [159:144] tile_dim2 = 0 (2D)
        g1[5] = HH * WW;                // [191:160] tensor_dim0_stride.lo = 262144 elems (plane stride)
        g1[6] = 0;                      // [207:192] stride.hi=0, [223:208] tensor_dim1_stride.lo=0
        g1[7] = 0;

        const i32x4 gz4 = {0, 0, 0, 0};             // groups 2/3 unused (<=2D tensor)
        const i32x8 gz8 = {0, 0, 0, 0, 0, 0, 0, 0}; // 6-arg toolchain's extra group, zero-filled
        __builtin_amdgcn_tensor_load_to_lds(g0, g1, gz4, gz4, gz8, /*cpol=*/0);
        __builtin_amdgcn_s_wait_tensorcnt(0);               // TENSORcnt -> 0: LDS tile resident
    }
    __syncthreads();                                        // publish LDS tile to all 8 waves
    asm volatile("" ::: "memory");                          // TDM wrote LDS behind the compiler's back

    // ---- butterfly: each thread handles 2 adjacent pixels (w = 2t, 2t+1) ----
    // LDS b64 reads, lane i hits banks {2i, 2i+1} mod 64 -> conflict-free.
    const int t  = threadIdx.x;
    const int w0 = 2 * t;

    const float a0 = tile[0 * WW + w0], a1 = tile[0 * WW + w0 + 1];
    const float h0 = tile[1 * WW + w0], h1 = tile[1 * WW + w0 + 1];
    const float v0 = tile[2 * WW + w0], v1 = tile[2 * WW + w0 + 1];
    const float d0 = tile[3 * WW + w0], d1 = tile[3 * WW + w0 + 1];

    // x00=(a+h+v+d)/2  x01=(a+h-v-d)/2  x10=(a-h+v-d)/2  x11=(a-h-v+d)/2
    const float p0 = a0 + d0, q0 = a0 - d0, r0s = h0 + v0, s0 = h0 - v0;
    const float p1 = a1 + d1, q1 = a1 - d1, r1s = h1 + v1, s1 = h1 - v1;

    const float x00_0 = (p0 + r0s) * 0.5f, x01_0 = (q0 + s0) * 0.5f;
    const float x10_0 = (q0 - s0) * 0.5f,  x11_0 = (p0 - r0s) * 0.5f;
    const float x00_1 = (p1 + r1s) * 0.5f, x01_1 = (q1 + s1) * 0.5f;
    const float x10_1 = (q1 - s1) * 0.5f,  x11_1 = (p1 - r1s) * 0.5f;

    // Interleave along W in registers -> one float4 per output row per thread.
    const f32x4 top = {x00_0, x01_0, x00_1, x01_1};   // row 2h,   cols 4t..4t+3
    const f32x4 bot = {x10_0, x11_0, x10_1, x11_1};   // row 2h+1, cols 4t..4t+3

    float* o0 = out + ((((size_t)b * 3 + c) * (2 * HH) + (size_t)(2 * h)) * (size_t)(2 * WW))
                    + (size_t)(4 * t);
    float* o1 = o0 + 2 * WW;

    // 403 MB streamed > 192 MB L2, never re-read -> non-temporal 128-bit stores,
    // 512 B contiguous per wave per instruction.
    __builtin_nontemporal_store(top, (f32x4*)o0);
    __builtin_nontemporal_store(bot, (f32x4*)o1);
}

extern "C" void kernel_launch(void* const* d_in, const int* in_sizes, int n_in,
                              void* d_out, int out_size, void* d_ws, size_t ws_size,
                              hipStream_t stream) {
    const float* in  = (const float*)d_in[0];   // (16, 12, 512, 512) f32
    float*       out = (float*)d_out;           // (16, 3, 1024, 1024) f32
    (void)in_sizes; (void)n_in; (void)out_size; (void)d_ws; (void)ws_size;

    const int blocks = B_ * 3 * HH;             // 24576 blocks, one (b,c,h) row each
    iwt_haar_tdm<<<blocks, 256, 0, stream>>>(in, out);
}